// LSTM_7499012898986
// MI455X (gfx1250) — compile-verified
//
#include <hip/hip_runtime.h>

typedef float v2f __attribute__((ext_vector_type(2)));
typedef float v8f __attribute__((ext_vector_type(8)));
typedef unsigned int u32x4 __attribute__((ext_vector_type(4)));
typedef int i32x4 __attribute__((ext_vector_type(4)));
typedef int i32x8 __attribute__((ext_vector_type(8)));

#define HIDDEN 32
#define FUTURE 6
#define BATCH  256
#define TLEN   2048
#define FEAT   64
#define GATES  128  // 4*HIDDEN

// ---------------------------------------------------------------------------
// Kernel 1: x_proj[B*T, 128] = X[B*T, 64] @ Wx[64, 128] + b
// fp32 WMMA (V_WMMA_F32_16X16X4_F32). One wave -> 16x128 strip.
// Wx staged into LDS once per block via the Tensor Data Mover (TDM),
// B fragments batched from LDS so WMMAs issue back-to-back.
// Block = 256 threads = 8 waves -> 128 rows/block; grid = B*T/128 = 4096.
// ---------------------------------------------------------------------------
__global__ __launch_bounds__(256)
void lstm_xproj_wmma(const float* __restrict__ X, const float* __restrict__ Wx,
                     const float* __restrict__ b, float* __restrict__ xproj) {
  __shared__ float sWx[FEAT * GATES];              // 64x128 fp32 = 32 KB

  const int lane = threadIdx.x & 31;
  const int wave = threadIdx.x >> 5;
  const int m15  = lane & 15;
  const int hi   = lane >> 4;                      // lanes 16-31 hold K+2
  const long row0 = (long)blockIdx.x * 128 + (long)wave * 16;

  // ---- TDM: DMA Wx (2-D tensor, 128 wide x 64 rows, fp32) into LDS --------
  if (wave == 0) {
    unsigned lds_off = (unsigned)(size_t)(void*)sWx;       // LDS byte offset
    unsigned long long ga = (unsigned long long)(size_t)Wx;
    u32x4 g0;
    g0[0] = 1u;                                            // count=1 (valid D#)
    g0[1] = lds_off;                                       // lds_addr
    g0[2] = (unsigned)(ga & 0xffffffffu);                  // global_addr[31:0]
    g0[3] = (unsigned)((ga >> 32) & 0x1ffffffu) | (2u << 30); // [56:32]|type=2
    i32x8 g1;
    g1[0] = (int)(2u << 16);                 // wg_mask=0, data_size=2 (4 B)
    g1[1] = (int)((GATES & 0xffffu) << 16);  // tensor_dim0[15:0]=128
    g1[2] = (int)((FEAT & 0xffffu) << 16);   // dim0 hi=0, tensor_dim1[15:0]=64
    g1[3] = (int)(GATES << 16);              // dim1 hi=0, tile_dim0=128
    g1[4] = (int)FEAT;                       // tile_dim1=64, tile_dim2=0
    g1[5] = (int)GATES;                      // tensor_dim0_stride=128
    g1[6] = 0;
    g1[7] = 0;
    i32x4 g2 = {0, 0, 0, 0};
    i32x4 g3 = {0, 0, 0, 0};
#if __clang_major__ >= 23
    i32x8 g4 = {0, 0, 0, 0, 0, 0, 0, 0};
    __builtin_amdgcn_tensor_load_to_lds(g0, g1, g2, g3, g4, 0);
#else
    __builtin_amdgcn_tensor_load_to_lds(g0, g1, g2, g3, 0);
#endif
    __builtin_amdgcn_s_wait_tensorcnt(0);    // data in LDS before barrier
  }

  // A fragments: 16 tiles of 16x4 fp32 covering K = 0..63 (overlaps TDM).
  // Layout (ISA 7.12.2): lane L<16 holds A[M=L][K=0,1]; lane L>=16 holds K=2,3.
  v2f a[16];
#pragma unroll
  for (int k4 = 0; k4 < 16; ++k4) {
    const float* p = X + (row0 + m15) * FEAT + k4 * 4 + hi * 2;
    a[k4].x = p[0];
    a[k4].y = p[1];
  }

  __syncthreads();                           // Wx visible in LDS to all waves

  // Double-buffered B fragments: all 16 frags of an N-tile loaded before the
  // 16-deep WMMA accumulation chain; next tile's ds_loads overlap WMMAs.
  v2f bf[2][16];
#pragma unroll
  for (int k4 = 0; k4 < 16; ++k4) {
    const int krow = k4 * 4 + hi * 2;
    bf[0][k4].x = sWx[(krow + 0) * GATES + m15];
    bf[0][k4].y = sWx[(krow + 1) * GATES + m15];
  }

#pragma unroll
  for (int nt = 0; nt < 8; ++nt) {
    const int n0 = nt * 16;
    const int cur = nt & 1;
    if (nt < 7) {
      const int n1 = n0 + 16;
#pragma unroll
      for (int k4 = 0; k4 < 16; ++k4) {
        const int krow = k4 * 4 + hi * 2;
        bf[cur ^ 1][k4].x = sWx[(krow + 0) * GATES + n1 + m15];
        bf[cur ^ 1][k4].y = sWx[(krow + 1) * GATES + n1 + m15];
      }
    }

    const float bias = b[n0 + m15];          // bias depends only on N
    v8f acc;
#pragma unroll
    for (int v = 0; v < 8; ++v) acc[v] = bias;

#pragma unroll
    for (int k4 = 0; k4 < 16; ++k4) {
      acc = __builtin_amdgcn_wmma_f32_16x16x4_f32(
          /*neg_a=*/false, a[k4], /*neg_b=*/false, bf[cur][k4],
          /*c_mod=*/(short)0, acc, /*reuse_a=*/false, /*reuse_b=*/false);
    }

    // C/D layout: VGPR v, lanes 0-15 -> M=v, N=lane; lanes 16-31 -> M=v+8.
#pragma unroll
    for (int v = 0; v < 8; ++v) {
      long m = row0 + v + hi * 8;
      xproj[m * GATES + n0 + m15] = acc[v];
    }
  }
}

// ---------------------------------------------------------------------------
// Kernel 2: sequential LSTM recurrence + fused elu/Dense head.
// One wave32 per batch row; lane j owns h[j], c[j] and the Wh columns
// {j, 32+j, 64+j, 96+j} (128 floats -> VGPRs). h broadcast via v_readlane.
// ---------------------------------------------------------------------------
__device__ __forceinline__ float fast_rcp(float x) { return __builtin_amdgcn_rcpf(x); }
__device__ __forceinline__ float fast_exp2(float x) { return __builtin_amdgcn_exp2f(x); }

__device__ __forceinline__ float sigmoid_fast(float z) {
  const float LOG2E = 1.4426950408889634f;
  return fast_rcp(1.0f + fast_exp2(-LOG2E * z));
}
__device__ __forceinline__ float tanh_fast(float z) {
  const float TWO_LOG2E = 2.8853900817779268f;
  float e = fast_exp2(TWO_LOG2E * z);              // e^{2z}
  return 1.0f - 2.0f * fast_rcp(e + 1.0f);         // (e-1)/(e+1)
}

__global__ __launch_bounds__(32)
void lstm_recur(const float* __restrict__ xproj, const float* __restrict__ Wh,
                const float* __restrict__ Wd, const float* __restrict__ bd,
                float* __restrict__ out) {
  const int j = threadIdx.x & 31;                  // hidden index
  const int bidx = blockIdx.x;                     // batch row

  // Per-lane slice of Wh held in registers: w[g][k] = Wh[k][g*32 + j]
  float w[4][HIDDEN];
#pragma unroll
  for (int g = 0; g < 4; ++g)
#pragma unroll
    for (int k = 0; k < HIDDEN; ++k)
      w[g][k] = Wh[k * GATES + g * HIDDEN + j];

  float h = 0.0f, c = 0.0f;
  const float* xp = xproj + (size_t)bidx * TLEN * GATES + j;

  for (int t = 0; t < TLEN; ++t) {
    const float* xr = xp + (size_t)t * GATES;
    // Issue loads early; latency hidden under the 160-op dot-product chain.
    float xi = xr[0], xf = xr[32], xg = xr[64], xo = xr[96];
    __builtin_prefetch(xr + GATES, 0, 1);          // global_prefetch next step

    float zi = 0.0f, zf = 0.0f, zg = 0.0f, zo = 0.0f;
#pragma unroll
    for (int k = 0; k < HIDDEN; ++k) {
      float hk = __builtin_bit_cast(
          float, __builtin_amdgcn_readlane(__builtin_bit_cast(int, h), k));
      zi = fmaf(hk, w[0][k], zi);
      zf = fmaf(hk, w[1][k], zf);
      zg = fmaf(hk, w[2][k], zg);
      zo = fmaf(hk, w[3][k], zo);
    }
    zi += xi; zf += xf; zg += xg; zo += xo;

    float gi = sigmoid_fast(zi);
    float gf = sigmoid_fast(zf);
    float gg = tanh_fast(zg);
    float go = sigmoid_fast(zo);

    c = fmaf(gf, c, gi * gg);
    h = go * tanh_fast(c);
  }

  // Fused head: out[b,:] = elu(h_T) @ Wd + bd
  const float LOG2E = 1.4426950408889634f;
  float eh = (h > 0.0f) ? h : (fast_exp2(LOG2E * h) - 1.0f);

  float s[FUTURE];
#pragma unroll
  for (int p = 0; p < FUTURE; ++p) s[p] = eh * Wd[j * FUTURE + p];
#pragma unroll
  for (int p = 0; p < FUTURE; ++p) {
#pragma unroll
    for (int off = 16; off >= 1; off >>= 1)
      s[p] += __shfl_xor(s[p], off, 32);
  }
  if (j == 0) {
#pragma unroll
    for (int p = 0; p < FUTURE; ++p) out[bidx * FUTURE + p] = s[p] + bd[p];
  }
}

// ---------------------------------------------------------------------------
extern "C" void kernel_launch(void* const* d_in, const int* in_sizes, int n_in,
                              void* d_out, int out_size, void* d_ws, size_t ws_size,
                              hipStream_t stream) {
  const float* X  = (const float*)d_in[0];   // [B,T,F]
  const float* Wx = (const float*)d_in[1];   // [F,4H]
  const float* Wh = (const float*)d_in[2];   // [H,4H]
  const float* b  = (const float*)d_in[3];   // [4H]
  const float* Wd = (const float*)d_in[4];   // [H,FUTURE]
  const float* bd = (const float*)d_in[5];   // [FUTURE]
  float* out   = (float*)d_out;              // [B,FUTURE]
  float* xproj = (float*)d_ws;               // [B*T,128] fp32 = 268 MB scratch

  lstm_xproj_wmma<<<dim3((BATCH * TLEN) / 128), dim3(256), 0, stream>>>(
      X, Wx, b, xproj);
  lstm_recur<<<dim3(BATCH), dim3(32), 0, stream>>>(xproj, Wh, Wd, bd, out);
}